// SpiralDeblock_53687091200507
// MI455X (gfx1250) — compile-verified
//
#include <hip/hip_runtime.h>
#include <hip/hip_bf16.h>

// Problem constants (from reference)
#define B_    8
#define N_IN  7056
#define N_UP  28224
#define SEQ   9
#define CIN   64
#define COUT  32
#define E_    (3 * N_UP)          // 84672
#define KDIM  (SEQ * CIN)         // 576
#define KSTEPS (KDIM / 4)         // 144 WMMA k-steps
#define ASTR  580                 // padded LDS row stride (580 % 64 == 4 -> conflict-free frag reads)
#define CAP   32                  // edges-per-row bucket capacity (rows ~ Poisson(3))

typedef __attribute__((ext_vector_type(2))) float v2f;
typedef __attribute__((ext_vector_type(8))) float v8f;

// ---------------- Kernel 1: zero per-row edge counts ----------------
__global__ void zero_cnt_kernel(int* __restrict__ cnt, int n) {
    int i = blockIdx.x * blockDim.x + threadIdx.x;
    if (i < n) cnt[i] = 0;
}

// ---------------- Kernel 2: repack W into B-fragment order ----------------
// wp[t*64 + acc*32 + lane] = { W[kk*32 + n], W[(kk+1)*32 + n] }
//   kk = 4*t + 2*(lane>>4), n = (lane&15) + 16*acc
__global__ void repack_w_kernel(const float* __restrict__ W, float2* __restrict__ wp) {
    int tid = blockIdx.x * blockDim.x + threadIdx.x;
    if (tid >= KSTEPS * 2 * 32) return;
    int lane = tid & 31;
    int acc  = (tid >> 5) & 1;
    int t    = tid >> 6;
    int kk   = 4 * t + 2 * (lane >> 4);
    int n    = (lane & 15) + 16 * acc;
    wp[tid] = make_float2(W[kk * COUT + n], W[(kk + 1) * COUT + n]);
}

// ---------------- Kernel 3: bucket edges by destination row ----------------
// Replaces 43M float atomics with E=84,672 int atomics.
__global__ void bucket_build_kernel(const int* __restrict__ row,
                                    int* __restrict__ cnt,
                                    int* __restrict__ bucket) {
    int e = blockIdx.x * blockDim.x + threadIdx.x;
    if (e >= E_) return;
    int r = row[e];
    int pos = atomicAdd(&cnt[r], 1);
    if (pos < CAP) bucket[r * CAP + pos] = e;
}

// ---------------- Kernel 4: pooled[b,r,:] = sum over incident edges (gather, no atomics) ----------------
// One thread per (b, r, channel-pair). Group of 32 threads shares (b, r):
// edge metadata loads are wave-uniform broadcasts; x reads are 256B coalesced.
__global__ void pool_rows_kernel(const float* __restrict__ x,
                                 const float* __restrict__ val,
                                 const int* __restrict__ col,
                                 const int* __restrict__ cnt,
                                 const int* __restrict__ bucket,
                                 float* __restrict__ pooled) {
    size_t gid = (size_t)blockIdx.x * blockDim.x + threadIdx.x;
    if (gid >= (size_t)B_ * N_UP * 32) return;
    int c2 = ((int)(gid & 31)) * 2;          // channel pair
    size_t grp = gid >> 5;
    int r = (int)(grp % N_UP);
    int b = (int)(grp / N_UP);

    int n = cnt[r];
    if (n > CAP) n = CAP;
    const int* bk = bucket + (size_t)r * CAP;
    const float* xb = x + (size_t)b * N_IN * CIN;

    float2 acc = make_float2(0.f, 0.f);
    // accumulate in ascending edge-id order -> bit-deterministic output
    int prev = -1;
    for (int i = 0; i < n; ++i) {
        int emin = 0x7fffffff;
        for (int j = 0; j < n; ++j) {
            int e = bk[j];
            if (e > prev && e < emin) emin = e;
        }
        prev = emin;
        float v  = val[emin];
        int   cl = col[emin];
        const float2 xx = *(const float2*)(xb + (size_t)cl * CIN + c2);
        acc.x += xx.x * v;
        acc.y += xx.y * v;
    }
    *(float2*)(pooled + ((size_t)b * N_UP + r) * CIN + c2) = acc;
}

// ---------------- Kernel 5: fused gather + WMMA GEMM + bias + relu ----------------
// One wave per (16-row tile of N_UP, batch). C tile = 16x32 via two f32 accumulators.
__global__ __launch_bounds__(32) void spiral_gemm_kernel(const float* __restrict__ pooled,
                                                         const float2* __restrict__ wp,
                                                         const float* __restrict__ bias,
                                                         const int* __restrict__ indices,
                                                         float* __restrict__ out) {
    __shared__ __align__(16) float aTile[16 * ASTR];   // 37,120 bytes

    const int n0   = blockIdx.x * 16;
    const int bb   = blockIdx.y;
    const int lane = threadIdx.x;

    // ---- Stage A tile: 16 rows x 576 cols gathered from pooled via indices ----
    const float* psrc = pooled + (size_t)bb * N_UP * CIN;
    const int c2 = lane * 2;                 // each lane moves 2 consecutive floats
    #pragma unroll 4
    for (int seg = 0; seg < 16 * SEQ; ++seg) {
        int m = seg / SEQ;
        int s = seg - m * SEQ;
        int idx = indices[(n0 + m) * SEQ + s];           // uniform across wave -> broadcast
        float2 v = *(const float2*)(psrc + (size_t)idx * CIN + c2);
        *(float2*)&aTile[m * ASTR + s * CIN + c2] = v;
    }
    __syncthreads();

    // ---- K loop: 144 steps of v_wmma_f32_16x16x4_f32, two N-tiles ----
    v8f c0 = {};
    v8f c1 = {};
    const int mA    = lane & 15;
    const int khalf = (lane >> 4) * 2;       // lanes 16-31 hold K = k0+2, k0+3
    const float* arow = &aTile[mA * ASTR + khalf];
    const v2f* wpv = (const v2f*)wp;

    #pragma unroll 4
    for (int t = 0; t < KSTEPS; ++t) {
        v2f a   = *(const v2f*)(arow + 4 * t);           // ds_load_2addr_b64, conflict-free
        v2f b0v = wpv[(t * 2 + 0) * 32 + lane];          // coalesced 256B/wave, L2-hot
        v2f b1v = wpv[(t * 2 + 1) * 32 + lane];
        c0 = __builtin_amdgcn_wmma_f32_16x16x4_f32(false, a, false, b0v, (short)0, c0, false, false);
        c1 = __builtin_amdgcn_wmma_f32_16x16x4_f32(false, a, false, b1v, (short)0, c1, false, false);
    }

    // ---- Epilogue: bias + relu + store (C layout: VGPR r -> M=r+8*(lane>>4), N=lane&15) ----
    const int ncol = lane & 15;
    const int mhi  = (lane >> 4) * 8;
    const float bias0 = bias[ncol];
    const float bias1 = bias[ncol + 16];
    #pragma unroll
    for (int r = 0; r < 8; ++r) {
        int m = r + mhi;
        size_t o = ((size_t)bb * N_UP + (n0 + m)) * COUT;
        out[o + ncol]      = fmaxf(c0[r] + bias0, 0.f);
        out[o + ncol + 16] = fmaxf(c1[r] + bias1, 0.f);
    }
}

extern "C" void kernel_launch(void* const* d_in, const int* in_sizes, int n_in,
                              void* d_out, int out_size, void* d_ws, size_t ws_size,
                              hipStream_t stream) {
    const float* x       = (const float*)d_in[0];
    const float* val     = (const float*)d_in[1];
    const float* W       = (const float*)d_in[2];
    const float* bias    = (const float*)d_in[3];
    const int*   row     = (const int*)d_in[4];
    const int*   col     = (const int*)d_in[5];
    const int*   indices = (const int*)d_in[6];
    float* out = (float*)d_out;

    // Workspace layout (all 16B aligned)
    const size_t pool_elems = (size_t)B_ * N_UP * CIN;        // 14,450,688 floats = 57.8 MB
    char* wsp = (char*)d_ws;
    float*  pooled = (float*)wsp;                              wsp += pool_elems * sizeof(float);
    float2* wp     = (float2*)wsp;                             wsp += (size_t)KSTEPS * 2 * 32 * sizeof(float2); // 73,728 B
    int*    cnt    = (int*)wsp;                                wsp += (size_t)N_UP * sizeof(int);               // 112,896 B
    int*    bucket = (int*)wsp;                                // N_UP * CAP * 4 = 3,612,672 B

    // 1) zero edge counts
    zero_cnt_kernel<<<(N_UP + 255) / 256, 256, 0, stream>>>(cnt, N_UP);
    // 2) repack W (tiny)
    {
        int total = KSTEPS * 2 * 32;                          // 9216
        repack_w_kernel<<<(total + 255) / 256, 256, 0, stream>>>(W, wp);
    }
    // 3) bucket edges by destination row (E int atomics)
    bucket_build_kernel<<<(E_ + 255) / 256, 256, 0, stream>>>(row, cnt, bucket);
    // 4) pooled = gather-accumulate (no float atomics, no zero-init of pooled)
    {
        size_t total = (size_t)B_ * N_UP * 32;                // 7,225,344
        int blocks = (int)((total + 255) / 256);
        pool_rows_kernel<<<blocks, 256, 0, stream>>>(x, val, col, cnt, bucket, pooled);
    }
    // 5) fused gather + WMMA GEMM
    {
        dim3 grid(N_UP / 16, B_);                             // (1764, 8)
        spiral_gemm_kernel<<<grid, 32, 0, stream>>>(pooled, wp, bias, indices, out);
    }
}